// Detector_68410239091278
// MI455X (gfx1250) — compile-verified
//
#include <hip/hip_runtime.h>
#include <stdint.h>
#include <math.h>

// ---- problem constants (match reference) ----
#define NBATCH   8
#define HWN      21760      // 128^2 + 64^2 + 32^2 + 16^2
#define NCLS     80
#define NCH      85         // 80 cls + 4 box + 1 nks
#define TOPKN    100
#define CONF_T   0.05f
#define NMS_T    0.5f
#define NKS_A    0.6f
// ---- algorithm parameters ----
#define NBUCKET  2048
#define CAPN     4096
#define ROWS     32

typedef unsigned int u32x4 __attribute__((ext_vector_type(4)));
typedef int          i32x4 __attribute__((ext_vector_type(4)));
typedef int          i32x8 __attribute__((ext_vector_type(8)));

// The TDM builtins only exist in the device pass; the host pass must fall through.
#if __has_builtin(__builtin_amdgcn_tensor_load_to_lds) && __has_builtin(__builtin_amdgcn_s_wait_tensorcnt)
#define USE_TDM 1
#else
#define USE_TDM 0
#if defined(__HIP_DEVICE_COMPILE__)
// Hard requirement on the DEVICE side: compile success proves tensor_load_to_lds
// + s_wait_tensorcnt are in the emitted gfx1250 code.
#error "gfx1250 device pass lacks tensor_load_to_lds / s_wait_tensorcnt builtins"
#endif
#endif

// DMA a [nrows x 85] fp32 tile (row stride 85 elements) from global into LDS via TDM.
__device__ __forceinline__ void tdm_load_tile(const float* gsrc, unsigned lds_byte_off, int nrows) {
#if USE_TDM
  unsigned long long ga = (unsigned long long)(uintptr_t)gsrc;
  u32x4 g0;
  g0.x = 1u;                                                 // count=1 valid user D#
  g0.y = lds_byte_off;                                       // lds_addr
  g0.z = (unsigned)(ga & 0xffffffffull);                     // global_addr[31:0]
  g0.w = (unsigned)((ga >> 32) & 0x01ffffffull) | (2u << 30);// global_addr[56:32] | type=2
  i32x8 g1;
  g1[0] = (int)(2u << 16);                 // data_size = 4B
  g1[1] = (int)((unsigned)NCH << 16);      // tensor_dim0 = 85 (bits 63:48)
  g1[2] = (int)((unsigned)HWN << 16);      // tensor_dim1 lo16 (bits 95:80); dim0 hi16 = 0
  g1[3] = (int)((unsigned)NCH << 16);      // tile_dim0 = 85 (bits 127:112); dim1 hi16 = 0
  g1[4] = (int)((unsigned)nrows & 0xffff); // tile_dim1 = rows (bits 143:128)
  g1[5] = NCH;                             // tensor_dim0_stride lo32 = 85
  g1[6] = 0;
  g1[7] = 0;
  i32x4 gz = {0, 0, 0, 0};
#if defined(__clang_major__) && __clang_major__ >= 23
  i32x8 gz8 = {0, 0, 0, 0, 0, 0, 0, 0};
  __builtin_amdgcn_tensor_load_to_lds(g0, g1, gz, gz, gz8, 0);
#else
  __builtin_amdgcn_tensor_load_to_lds(g0, g1, gz, gz, 0);
#endif
  __builtin_amdgcn_s_wait_tensorcnt(0);
#else
  (void)gsrc; (void)lds_byte_off; (void)nrows;
#endif
}

__global__ __launch_bounds__(256) void init_kernel(unsigned* p, int n) {
  int i = blockIdx.x * blockDim.x + threadIdx.x;
  if (i < n) p[i] = 0u;
}

// mode 0: build per-batch value histogram of local peaks.
// mode 1: compact peaks with bucket >= thr[b] into candidate lists.
__global__ __launch_bounds__(256) void pass_kernel(
    const float* __restrict__ pred, unsigned* __restrict__ hist,
    const unsigned* __restrict__ thr, unsigned* __restrict__ cnt,
    float* __restrict__ cval, unsigned* __restrict__ ckey, int mode) {
  __shared__ float    raw[(ROWS + 2) * NCH];
  __shared__ float    ts[ROWS + 2][NCLS + 2];
  __shared__ float    ex[ROWS + 2];
  __shared__ unsigned hist_s[NBUCKET];

  const int b   = blockIdx.y;
  const int i0  = blockIdx.x * ROWS;
  const int start = (i0 > 0) ? (i0 - 1) : 0;
  int endr = i0 + ROWS + 1; if (endr > HWN) endr = HWN;
  const int nrows = endr - start;
  const int tid = threadIdx.x;

  if (mode == 0)
    for (int i = tid; i < NBUCKET; i += blockDim.x) hist_s[i] = 0u;

  const float* gsrc = pred + ((size_t)b * HWN + start) * NCH;
  if ((tid >> 5) == 0) {  // wave 0 only issues the TDM op (EXEC-independent, per-wave issue)
    tdm_load_tile(gsrc, (unsigned)(uintptr_t)&raw[0], nrows);
  }
  __syncthreads();

  // per-row exponent from nks channel (channel 84), plus class halo init
  if (tid < ROWS + 2) {
    int g = i0 - 1 + tid;
    float e = 1.0f;
    if (g >= 0 && g < HWN) {
      float y   = raw[(g - start) * NCH + (NCH - 1)];
      float n0  = 1.0f / (1.0f + __expf(-y));
      float nks = 1.0f / (1.0f + __expf(-2.0f * n0 + 1.0f));
      e = (2.0f - nks) * NKS_A + 1e-14f;
    }
    ex[tid] = e;
    ts[tid][0]        = -INFINITY;
    ts[tid][NCLS + 1] = -INFINITY;
  }
  __syncthreads();

  // transformed class scores: sigmoid(x)^e
  for (int idx = tid; idx < (ROWS + 2) * NCLS; idx += blockDim.x) {
    int rr = idx / NCLS, c = idx % NCLS;
    int g = i0 - 1 + rr;
    float t = -INFINITY;
    if (g >= 0 && g < HWN) {
      float x = raw[(g - start) * NCH + c];
      float s = 1.0f / (1.0f + __expf(-x));
      t = __powf(s, ex[rr]);
    }
    ts[rr][c + 1] = t;
  }
  __syncthreads();

  const int thr_b = (mode == 1) ? (int)thr[b] : 0;

  // 3x3 (flat-HW x class) peak test, exactly mirroring the reference pool semantics
  for (int idx = tid; idx < ROWS * NCLS; idx += blockDim.x) {
    int rr = idx / NCLS + 1, c = idx % NCLS;
    int g = i0 + rr - 1;
    if (g >= HWN) continue;
    float t = ts[rr][c + 1];
    float m = ts[rr - 1][c];
    m = fmaxf(m, ts[rr - 1][c + 1]);
    m = fmaxf(m, ts[rr - 1][c + 2]);
    m = fmaxf(m, ts[rr][c]);
    m = fmaxf(m, ts[rr][c + 2]);
    m = fmaxf(m, ts[rr + 1][c]);
    m = fmaxf(m, ts[rr + 1][c + 1]);
    m = fmaxf(m, ts[rr + 1][c + 2]);
    if (t >= m) {  // hmax == cmap (t is in window, so t >= neighbors <=> t == window max)
      int bkt = (int)(t * (float)NBUCKET);
      if (bkt >= NBUCKET) bkt = NBUCKET - 1;
      if (bkt < 0) bkt = 0;
      if (mode == 0) {
        atomicAdd(&hist_s[bkt], 1u);
      } else if (bkt >= thr_b) {
        unsigned pos = atomicAdd(&cnt[b], 1u);
        if (pos < CAPN) {
          cval[b * CAPN + pos] = t;
          ckey[b * CAPN + pos] = (unsigned)(c * HWN + g);  // class-major flat index
        }
      }
    }
  }

  if (mode == 0) {
    __syncthreads();
    for (int i = tid; i < NBUCKET; i += blockDim.x)
      if (hist_s[i]) atomicAdd(&hist[b * NBUCKET + i], hist_s[i]);
  }
}

// find largest bucket whose suffix-count >= TOPK (per batch)
__global__ void scan_kernel(const unsigned* __restrict__ hist, unsigned* __restrict__ thr) {
  int b = threadIdx.x;
  if (b >= NBATCH) return;
  unsigned acc = 0; int t = 0;
  for (int i = NBUCKET - 1; i >= 0; --i) {
    acc += hist[b * NBUCKET + i];
    if (acc >= TOPKN) { t = i; break; }
  }
  thr[b] = (unsigned)t;
}

// one block per batch: bitonic sort candidates, take top-100, decode boxes, NMS, write outputs
__global__ __launch_bounds__(256) void final_kernel(
    const float* __restrict__ pred, const float* __restrict__ pix,
    const unsigned* __restrict__ cnt, const float* __restrict__ cval,
    const unsigned* __restrict__ ckey, float* __restrict__ out) {
  __shared__ float    v_s[CAPN];
  __shared__ unsigned k_s[CAPN];
  __shared__ float    bx[TOPKN][4];
  __shared__ float    area[TOPKN];
  __shared__ int      cls_s[TOPKN];
  __shared__ float    sc[TOPKN];
  __shared__ int      keep_s[TOPKN];
  __shared__ int      sup[TOPKN];

  const int b = blockIdx.x;
  const int tid = threadIdx.x;
  int n = (int)cnt[b]; if (n > CAPN) n = CAPN;

  for (int i = tid; i < CAPN; i += blockDim.x) {
    if (i < n) { v_s[i] = cval[b * CAPN + i]; k_s[i] = ckey[b * CAPN + i]; }
    else       { v_s[i] = 0.0f;               k_s[i] = 0xffffffffu; }
  }
  __syncthreads();

  // Prefetch the scattered box-channel rows for likely top candidates while we sort.
  // (100 random ~16B gathers/batch is the latency-bound part; lowers to global_prefetch_b8.)
  if (tid < TOPKN && tid < n) {
    unsigned key = k_s[tid];
    unsigned hw = key % (unsigned)HWN;
    __builtin_prefetch(pred + ((size_t)b * HWN + hw) * NCH + NCLS, 0, 0);
    __builtin_prefetch(pix + (size_t)hw * 4, 0, 0);
  }

  // bitonic sort: descending by value, ties by ascending key (matches top_k tie-break)
  for (unsigned k = 2; k <= CAPN; k <<= 1) {
    for (unsigned j = k >> 1; j > 0; j >>= 1) {
      __syncthreads();
      for (unsigned i = tid; i < CAPN; i += blockDim.x) {
        unsigned ixj = i ^ j;
        if (ixj > i) {
          float vi = v_s[i], vx = v_s[ixj];
          unsigned ki = k_s[i], kx = k_s[ixj];
          bool prec_x = (vx > vi) || (vx == vi && kx < ki);  // ixj-elem should precede i-elem
          bool dir = ((i & k) == 0);
          if (prec_x == dir) {
            v_s[i] = vx; v_s[ixj] = vi;
            k_s[i] = kx; k_s[ixj] = ki;
          }
        }
      }
    }
  }
  __syncthreads();

  // out layout (floats): [0,3200) bboxes  [3200,4000) scores  [4000,4800) cls  [4800,5600) keep
  if (tid < TOPKN) {
    float s = v_s[tid];
    unsigned key = k_s[tid];
    int cls = 0, hw = 0;
    float box[4] = {0.f, 0.f, 0.f, 0.f};
    if (key != 0xffffffffu) {
      cls = (int)(key / (unsigned)HWN);
      hw  = (int)(key % (unsigned)HWN);
      const float* pp = pred + ((size_t)b * HWN + hw) * NCH + NCLS;
      const float w4[4] = {-1.f, -1.f, 1.f, 1.f};
      for (int j = 0; j < 4; ++j)
        box[j] = fmaxf(pp[j], 0.f) * w4[j] + pix[hw * 4 + j];
    }
    sc[tid] = s; cls_s[tid] = cls;
    for (int j = 0; j < 4; ++j) bx[tid][j] = box[j];
    area[tid] = (box[2] - box[0]) * (box[3] - box[1]);
    sup[tid] = 0; keep_s[tid] = 0;

    out[3200 + b * TOPKN + tid] = s;
    out[4000 + b * TOPKN + tid] = (float)cls;
    for (int j = 0; j < 4; ++j) {
      float v = box[j] * (1.0f / 512.0f);
      v = fminf(fmaxf(v, 0.0f), 1.0f);
      out[(b * TOPKN + tid) * 4 + j] = v;
    }
  }
  __syncthreads();

  // greedy class-aware NMS over the sorted 100 (sorted order == reference's argsort order)
  for (int i = 0; i < TOPKN; ++i) {
    if (tid == 0) keep_s[i] = (sc[i] > CONF_T) && !sup[i];
    __syncthreads();
    if (tid < TOPKN && tid > i && keep_s[i] && cls_s[tid] == cls_s[i]) {
      float xx1 = fmaxf(bx[i][0], bx[tid][0]);
      float yy1 = fmaxf(bx[i][1], bx[tid][1]);
      float xx2 = fminf(bx[i][2], bx[tid][2]);
      float yy2 = fminf(bx[i][3], bx[tid][3]);
      float w = fmaxf(1e-28f, xx2 - xx1);
      float h = fmaxf(1e-28f, yy2 - yy1);
      float inter = w * h;
      float ovr = inter / (area[i] + area[tid] - inter);
      if (ovr > NMS_T) sup[tid] = 1;
    }
    __syncthreads();
  }
  if (tid < TOPKN) out[4800 + b * TOPKN + tid] = keep_s[tid] ? 1.0f : 0.0f;
}

extern "C" void kernel_launch(void* const* d_in, const int* in_sizes, int n_in,
                              void* d_out, int out_size, void* d_ws, size_t ws_size,
                              hipStream_t stream) {
  (void)in_sizes; (void)n_in; (void)out_size; (void)ws_size;
  const float* pred = (const float*)d_in[0];      // (8, 21760, 85) fp32
  const float* pix  = (const float*)d_in[1];      // (21760, 4) fp32
  float* out = (float*)d_out;

  char* ws = (char*)d_ws;
  unsigned* hist = (unsigned*)(ws + 0);                    // 8*2048 u32
  unsigned* cnt  = (unsigned*)(ws + 65536);                // 8 u32
  unsigned* thr  = (unsigned*)(ws + 65568);                // 8 u32
  float*    cval = (float*)(ws + 65600);                   // 8*4096 f32
  unsigned* ckey = (unsigned*)(ws + 65600 + NBATCH * CAPN * 4);

  const int init_n = NBATCH * NBUCKET + 2 * NBATCH;        // hist + cnt + thr
  init_kernel<<<(init_n + 255) / 256, 256, 0, stream>>>((unsigned*)ws, init_n);

  dim3 pg((HWN + ROWS - 1) / ROWS, NBATCH);
  pass_kernel<<<pg, 256, 0, stream>>>(pred, hist, thr, cnt, cval, ckey, 0);
  scan_kernel<<<1, 32, 0, stream>>>(hist, thr);
  pass_kernel<<<pg, 256, 0, stream>>>(pred, hist, thr, cnt, cval, ckey, 1);
  final_kernel<<<NBATCH, 256, 0, stream>>>(pred, pix, cnt, cval, ckey, out);
}